// EGConv_74474732912710
// MI455X (gfx1250) — compile-verified
//
#include <hip/hip_runtime.h>

typedef float v2f __attribute__((ext_vector_type(2)));
typedef float v8f __attribute__((ext_vector_type(8)));

#define D_NODE 64
#define D_EDGE 32
#define D_IN   160
#define D_OUT  64
#define W1T_STRIDE 164   // padded K-stride for transposed W1 in LDS (bank-conflict-free b64 reads)
#define W2T_STRIDE 68
#define H_STRIDE   68
#define WAVES_PER_BLOCK 8
#define TILE_M 16

__global__ void egconv_zero_kernel(float* __restrict__ out, int n) {
    int i = blockIdx.x * blockDim.x + threadIdx.x;
    if (i < n) out[i] = 0.0f;
}

__global__ __launch_bounds__(256)
void egconv_mlp_scatter_kernel(const float* __restrict__ node_feats,
                               const float* __restrict__ edge_feats,
                               const long long* __restrict__ src,
                               const long long* __restrict__ dst,
                               const float* __restrict__ W1,
                               const float* __restrict__ b1,
                               const float* __restrict__ W2,
                               const float* __restrict__ b2,
                               float* __restrict__ out,
                               int n_edges)
{
    __shared__ float ldsW1t[D_OUT * W1T_STRIDE];                    // W1^T : [col][k], 160 k's
    __shared__ float ldsW2t[D_OUT * W2T_STRIDE];                    // W2^T : [col][k], 64 k's
    __shared__ float ldsH[WAVES_PER_BLOCK * TILE_M * H_STRIDE];     // per-wave h tile [row][k]

    const int tid = threadIdx.x;

    // Cooperative transposed load of weights into LDS (coalesced global reads).
    for (int idx = tid; idx < D_IN * D_OUT; idx += 256) {
        int k = idx >> 6;       // 0..159
        int c = idx & 63;       // 0..63
        ldsW1t[c * W1T_STRIDE + k] = W1[idx];
    }
    for (int idx = tid; idx < D_OUT * D_OUT; idx += 256) {
        int k = idx >> 6;
        int c = idx & 63;
        ldsW2t[c * W2T_STRIDE + k] = W2[idx];
    }
    __syncthreads();

    const int lane  = tid & 31;
    const int wave  = tid >> 5;
    const int row   = lane & 15;           // A-layout: M = lane % 16
    const int khalf = (lane >> 4) << 1;    // A/B layout: upper half-lanes take K+2,K+3
    const int colh  = lane & 15;           // B/C/D layout: N = lane % 16
    const int mhi   = (lane >> 4) << 3;    // C/D layout: upper half-lanes hold M+8

    const int tile = blockIdx.x * WAVES_PER_BLOCK + wave;
    const long long ebase = (long long)tile * TILE_M;

    long long e = ebase + row;
    long long eclamp = (e < n_edges) ? e : (long long)(n_edges - 1);

    const long long s = src[eclamp];
    const long long d = dst[eclamp];
    const float* __restrict__ psrc  = node_feats + s * D_NODE;   // x[:,   0: 64]
    const float* __restrict__ pdstf = node_feats + d * D_NODE;   // x[:,  64:128]
    const float* __restrict__ pedge = edge_feats + eclamp * D_EDGE; // x[:,128:160]

    // ---------- Layer 1: [16 x 160] @ [160 x 64], bias folded into C init ----------
    v8f acc[4];
#pragma unroll
    for (int n = 0; n < 4; ++n) {
        float bv = b1[n * 16 + colh];
        acc[n] = (v8f){bv, bv, bv, bv, bv, bv, bv, bv};
    }

#pragma unroll
    for (int kk = 0; kk < 40; ++kk) {
        const int k0 = kk * 4;
        const float* abase;
        int krel;
        if (k0 < 64)        { abase = psrc;  krel = k0;       }
        else if (k0 < 128)  { abase = pdstf; krel = k0 - 64;  }
        else                { abase = pedge; krel = k0 - 128; }
        v2f a = *(const v2f*)(abase + krel + khalf);
#pragma unroll
        for (int n = 0; n < 4; ++n) {
            const int col = n * 16 + colh;
            v2f b = *(const v2f*)(&ldsW1t[col * W1T_STRIDE + k0 + khalf]);
            acc[n] = __builtin_amdgcn_wmma_f32_16x16x4_f32(
                false, a, false, b, (short)0, acc[n], false, false);
        }
    }

    // ---------- ReLU + C->A relayout through LDS ----------
    float* hbuf = &ldsH[wave * TILE_M * H_STRIDE];
#pragma unroll
    for (int n = 0; n < 4; ++n) {
#pragma unroll
        for (int v = 0; v < 8; ++v) {
            float h = acc[n][v];
            h = h > 0.0f ? h : 0.0f;
            hbuf[(v + mhi) * H_STRIDE + n * 16 + colh] = h;
        }
    }
    // Same-wave LDS store->load is in-order (DScnt); compiler inserts s_wait_dscnt.

    // ---------- Layer 2: [16 x 64] @ [64 x 64] ----------
    v8f acc2[4];
#pragma unroll
    for (int n = 0; n < 4; ++n) {
        float bv = b2[n * 16 + colh];
        acc2[n] = (v8f){bv, bv, bv, bv, bv, bv, bv, bv};
    }

#pragma unroll
    for (int kk = 0; kk < 16; ++kk) {
        const int k0 = kk * 4;
        v2f a = *(const v2f*)(&hbuf[row * H_STRIDE + k0 + khalf]);
#pragma unroll
        for (int n = 0; n < 4; ++n) {
            const int col = n * 16 + colh;
            v2f b = *(const v2f*)(&ldsW2t[col * W2T_STRIDE + k0 + khalf]);
            acc2[n] = __builtin_amdgcn_wmma_f32_16x16x4_f32(
                false, a, false, b, (short)0, acc2[n], false, false);
        }
    }

    // ---------- Scatter-add messages to destination nodes ----------
#pragma unroll
    for (int v = 0; v < 8; ++v) {
        const int m = v + mhi;              // C/D-layout row held by this lane/vgpr
        const long long em = ebase + m;
        if (em < n_edges) {
            const long long dm = dst[em];
            float* orow = out + dm * D_OUT;
#pragma unroll
            for (int n = 0; n < 4; ++n) {
                atomicAdd(orow + n * 16 + colh, acc2[n][v]);
            }
        }
    }
}

extern "C" void kernel_launch(void* const* d_in, const int* in_sizes, int n_in,
                              void* d_out, int out_size, void* d_ws, size_t ws_size,
                              hipStream_t stream) {
    const float*     node_feats = (const float*)d_in[0];
    const float*     edge_feats = (const float*)d_in[1];
    const long long* src        = (const long long*)d_in[2];
    const long long* dst        = (const long long*)d_in[3];
    const float*     W1         = (const float*)d_in[4];
    const float*     b1         = (const float*)d_in[5];
    const float*     W2         = (const float*)d_in[6];
    const float*     b2         = (const float*)d_in[7];
    float* out = (float*)d_out;

    const int n_edges = in_sizes[2];

    // Zero-init the output accumulator (harness poisons d_out).
    egconv_zero_kernel<<<(out_size + 255) / 256, 256, 0, stream>>>(out, out_size);

    if (n_edges <= 0) return;
    const int tiles  = (n_edges + TILE_M - 1) / TILE_M;
    const int blocks = (tiles + WAVES_PER_BLOCK - 1) / WAVES_PER_BLOCK;
    egconv_mlp_scatter_kernel<<<blocks, 32 * WAVES_PER_BLOCK, 0, stream>>>(
        node_feats, edge_feats, src, dst, W1, b1, W2, b2, out, n_edges);
}